// Pool_86629490360848
// MI455X (gfx1250) — compile-verified
//
#include <hip/hip_runtime.h>

// Sparse 3D max-pool (torchsparse-style) for MI455X / gfx1250, wave32.
//
//   1) init_table_kernel : clear open-addressing hash table in d_ws
//   2) build_table_kernel: insert packed(coords) -> row index (keys unique)
//   3) pool_kernel       : one wave32 per output row.
//        phase 1: lanes 0..26 probe the 27 offset keys (int2 slot = one
//                 global_load_b64; table is L2-resident), prefetch hits.
//        phase 2: compact matched rows (ballot+rank), then DMA them into an
//                 LDS staging buffer with global_load_async_to_lds_b128
//                 (2 rows / instruction: half-wave each, 512B per issue),
//                 tracked by ASYNCcnt -> s_wait_asynccnt 0.
//        phase 3: per-lane float2 running max over staged rows from LDS.

#define EMPTY_KEY   (-1)
#define C_CHANNELS  64
#define POOL_WAVES  4                       // waves (= rows) per block
#define POOL_TB     (POOL_WAVES * 32)

static __device__ __forceinline__ int pack_key(int x, int y, int z, int b) {
    const int R = 258;  // S + 2; max key ~34.3M < 2^31
    return ((b * R + (x + 1)) * R + (y + 1)) * R + (z + 1);
}

static __device__ __forceinline__ unsigned mix_hash(unsigned k) {
    k ^= k >> 16; k *= 0x85ebca6bu;
    k ^= k >> 13; k *= 0xc2b2ae35u;
    k ^= k >> 16;
    return k;
}

__global__ void init_table_kernel(int2* __restrict__ table, int cap) {
    int i = blockIdx.x * blockDim.x + threadIdx.x;
    if (i < cap) table[i] = make_int2(EMPTY_KEY, -1);
}

__global__ void build_table_kernel(const int* __restrict__ coords, int n,
                                   int2* __restrict__ table, unsigned mask) {
    int i = blockIdx.x * blockDim.x + threadIdx.x;
    if (i >= n) return;
    const int4 c = ((const int4*)coords)[i];      // [x, y, z, batch]
    const int key = pack_key(c.x, c.y, c.z, c.w);
    unsigned slot = mix_hash((unsigned)key) & mask;
    int* tp = (int*)table;
    for (;;) {
        int old = atomicCAS(tp + 2 * (size_t)slot, EMPTY_KEY, key);
        if (old == EMPTY_KEY || old == key) {     // keys unique: first CAS wins
            tp[2 * (size_t)slot + 1] = i;
            return;
        }
        slot = (slot + 1) & mask;
    }
}

__global__ __launch_bounds__(POOL_TB) void pool_kernel(
        const float* __restrict__ feats,
        const int* __restrict__ down,
        const int2* __restrict__ table,
        unsigned mask, int m,
        float* __restrict__ out) {
    // per-wave staging: 27 feature rows of 256B, async-DMA'd from global
    __shared__ __align__(16) float stage[POOL_WAVES][27][C_CHANNELS];
    __shared__ int slist[POOL_WAVES][28];

    const int lane = threadIdx.x & 31;                       // wave32
    const int wave = threadIdx.x >> 5;
    const int row  = blockIdx.x * POOL_WAVES + wave;
    if (row >= m) return;

    const int4 d = ((const int4*)down)[row];

    // ---- phase 1: lanes 0..26 each resolve one kernel offset ----
    int src = -1;
    if (lane < 27) {
        const int dx = lane % 3 - 1;          // reference order: x fastest
        const int dy = (lane / 3) % 3 - 1;    // (order irrelevant for max)
        const int dz = lane / 9 - 1;
        const int key = pack_key(d.x + dx, d.y + dy, d.z + dz, d.w);
        unsigned slot = mix_hash((unsigned)key) & mask;
        for (;;) {
            const int2 e = table[slot];                      // b64 probe, L2-hot
            if (e.x == key)       { src = e.y; break; }
            if (e.x == EMPTY_KEY) break;
            slot = (slot + 1) & mask;
        }
        if (src >= 0) {
            // gfx1250 global_prefetch_b8: start warming the 256B feature row
            __builtin_prefetch(feats + (size_t)src * C_CHANNELS, 0, 1);
        }
    }

    // ---- phase 2: compact matches, async-DMA rows into LDS ----
    const unsigned hit  = (unsigned)__ballot(src >= 0);      // uniform
    const int      cnt  = __popc(hit);
    if (src >= 0) {
        const int rank = __popc(hit & ((1u << lane) - 1u));
        slist[wave][rank] = src;        // DS ops are in-order within a wave
    }
    // 2 rows per async instruction: lanes 0-15 -> row t, lanes 16-31 -> row t+1
    for (int t = 0; t < cnt; t += 2) {
        const int idx = t + (lane >> 4);
        if (idx < cnt) {
            const int    s  = slist[wave][idx];
            const int    l  = lane & 15;                     // 16B chunk id
            const float* gp = feats + (size_t)s * C_CHANNELS + l * 4;
            const unsigned lo =
                (unsigned)(size_t)&stage[wave][idx][l * 4];  // LDS byte offset
            asm volatile("global_load_async_to_lds_b128 %0, %1, off"
                         :: "v"(lo), "v"(gp) : "memory");
        }
    }
    if (cnt > 0) {
        asm volatile("s_wait_asynccnt 0x0" ::: "memory");    // ASYNCcnt drain
    }

    // ---- phase 3: per-lane float2 running max over staged rows ----
    float accx = -__builtin_inff();
    float accy = -__builtin_inff();
    const int c0 = lane * 2;
    for (int t = 0; t < cnt; ++t) {
        const float2 v = *(const float2*)&stage[wave][t][c0];
        accx = fmaxf(accx, v.x);
        accy = fmaxf(accy, v.y);
    }
    // mirror reference: any infinite result (incl. no-match -inf) -> 0
    accx = (__builtin_fabsf(accx) == __builtin_inff()) ? 0.0f : accx;
    accy = (__builtin_fabsf(accy) == __builtin_inff()) ? 0.0f : accy;
    *(float2*)(out + (size_t)row * C_CHANNELS + c0) = make_float2(accx, accy);
}

extern "C" void kernel_launch(void* const* d_in, const int* in_sizes, int n_in,
                              void* d_out, int out_size, void* d_ws, size_t ws_size,
                              hipStream_t stream) {
    const float* feats  = (const float*)d_in[0];
    const int*   coords = (const int*)d_in[1];
    const int*   down   = (const int*)d_in[2];
    float*       out    = (float*)d_out;

    const int N = in_sizes[0] / C_CHANNELS;   // input voxels
    const int M = in_sizes[2] / 4;            // output voxels

    // Hash table: power-of-two capacity, low load factor, L2-resident.
    int lg = 21;                               // 16 MB floor (2^21 int2 slots)
    while (lg < 24 &&
           ((size_t)sizeof(int2) << (lg + 1)) <= ws_size &&
           ((size_t)1 << lg) < (size_t)N * 4) {
        ++lg;
    }
    const int      cap  = 1 << lg;
    const unsigned mask = (unsigned)(cap - 1);
    int2* table = (int2*)d_ws;

    const int TB = 256;
    init_table_kernel<<<(cap + TB - 1) / TB, TB, 0, stream>>>(table, cap);
    build_table_kernel<<<(N + TB - 1) / TB, TB, 0, stream>>>(coords, N, table, mask);

    const int blocks = (M + POOL_WAVES - 1) / POOL_WAVES;
    pool_kernel<<<blocks, POOL_TB, 0, stream>>>(feats, down, table, mask, M, out);
}